// EPGSimulationMT_GPU_74990128988302
// MI455X (gfx1250) — compile-verified
//
#include <hip/hip_runtime.h>
#include <cstdint>

#define N_PULSES 4096
#define N_STATES 128

typedef uint32_t v4u __attribute__((ext_vector_type(4)));
typedef int      v8i __attribute__((ext_vector_type(8)));
typedef int      v4i __attribute__((ext_vector_type(4)));

// One TDM store: 2-D tile, X = 256 contiguous 8B elements (2 pulses x 128 states),
// Y = 6 rows (the six output arrays), global row stride = 4096*128 elements = 4 MB.
// LDS side is fully contiguous (stg[buf][arr][pulse][state]).
__device__ __forceinline__ void tdm_store_batch(uint32_t lds_addr, uint64_t gaddr) {
  v4u g0;
  g0[0] = 1u;                                                     // count=1 (valid D#)
  g0[1] = lds_addr;                                               // LDS byte offset
  g0[2] = (uint32_t)gaddr;                                        // global_addr[31:0]
  g0[3] = (uint32_t)((gaddr >> 32) & 0x01FFFFFFull) | (2u << 30); // addr[56:32] | type=2
  v8i g1;
  g1[0] = (int)(3u << 16);          // workgroup_mask=0, data_size=3 (8 bytes)
  g1[1] = (int)(256u << 16);        // tensor_dim0 = 256 elements
  g1[2] = (int)(6u  << 16);         // tensor_dim1 = 6 rows
  g1[3] = (int)(256u << 16);        // tile_dim0 = 256
  g1[4] = 6;                        // tile_dim1 = 6, tile_dim2 = 0
  g1[5] = (int)(4096u * 128u);      // tensor_dim0_stride = 524288 elems (4 MB/row)
  g1[6] = 0;
  g1[7] = 0;
  // 6-arg toolchain variant: (g0, g1, g2, g3, g4, cpol)
  __builtin_amdgcn_tensor_store_from_lds(g0, g1, (v4i)0, (v4i)0, (v8i)0, 0);
}

__global__ __launch_bounds__(128, 1)
void epg_mt_kernel(const float* __restrict__ fa_arr,
                   const float* __restrict__ ph_arr,
                   const float* __restrict__ pT1f, const float* __restrict__ pT2f,
                   const float* __restrict__ pT1b, const float* __restrict__ pT2b,
                   const float* __restrict__ pkf,  const float* __restrict__ pkb,
                   const float* __restrict__ pTR,  const float* __restrict__ pB0,
                   const float* __restrict__ pB1,  const float* __restrict__ pwf,
                   const float* __restrict__ pwb,
                   float* __restrict__ out) {
  // Triple-buffered output staging: 3 x 6 arrays x 2 pulses x 128 states x 8B = 36 KB
  __shared__ float2 stg[3][6][2][N_STATES];
  // Double-buffered roll-exchange buffer: 2 x 4 x 128 x 8B = 8 KB
  __shared__ float2 rollb[2][4][N_STATES];

  const int s = threadIdx.x;

  // ---- uniform scalar constants (s_loads) ----
  const float T1f = *pT1f, T2f = *pT2f, T1b = *pT1b, T2b = *pT2b;
  const float kf = *pkf, kb = *pkb, TR = *pTR, B0 = *pB0, B1 = *pB1;
  const float wf = *pwf, wb = *pwb;

  const float E1f = __expf(-TR / T1f);
  const float E2f = __expf(-TR / T2f);
  const float E1b = __expf(-TR / T1b);
  const float E2b = __expf(-TR / T2b);
  const float scale = TR * 0.001f;
  const float phi = 6.2831853071795864f * B0 * TR * 0.001f;
  const float rc = __cosf(phi), rs = __sinf(phi);    // rot = e^{i phi}
  const float rZf = (1.0f - E1f) * wf;
  const float rZb = (1.0f - E1b) * wb;

  // ---- initial carry ----
  float2 Fpf = {0.f, 0.f}, Fmf = {0.f, 0.f}, Fpb = {0.f, 0.f}, Fmb = {0.f, 0.f};
  float2 Zf  = {(s == 0) ? wf : 0.f, 0.f};
  float2 Zb  = {(s == 0) ? wb : 0.f, 0.f};

  const bool wave0 = (threadIdx.x < 32);     // wave-uniform predicate
  const uint64_t outb = (uint64_t)(uintptr_t)out;

  // software-pipelined uniform pulse loads
  float fa_n = fa_arr[0];
  float ph_n = ph_arr[0];

  for (int p = 0; p < N_PULSES; ++p) {
    const float fa = fa_n, ph = ph_n;
    const int pn = (p + 1 < N_PULSES) ? (p + 1) : p;
    fa_n = fa_arr[pn];
    ph_n = ph_arr[pn];

    // Gate staging-buffer reuse: with 3 buffers, allowing 1 outstanding TDM
    // guarantees the DMA that read this buffer (3 batches ago) is done.
    if (wave0 && (p & 1) == 0) __builtin_amdgcn_s_wait_tensorcnt(1);

    // ---- relaxation + exchange ----
    Fpf.x *= E2f; Fpf.y *= E2f;  Fmf.x *= E2f; Fmf.y *= E2f;
    Fpb.x *= E2b; Fpb.y *= E2b;  Fmb.x *= E2b; Fmb.y *= E2b;
    const float dZx = kb * Zb.x - kf * Zf.x;
    const float dZy = kb * Zb.y - kf * Zf.y;
    float2 Zf2 = {E1f * Zf.x + rZf + scale * dZx, E1f * Zf.y + scale * dZy};
    float2 Zb2 = {E1b * Zb.x + rZb - scale * dZx, E1b * Zb.y - scale * dZy};
    Zf = Zf2; Zb = Zb2;

    // ---- B0 dephasing: Fp *= rot, Fm *= conj(rot) ----
    {
      float2 t;
      t = {Fpf.x * rc - Fpf.y * rs, Fpf.x * rs + Fpf.y * rc}; Fpf = t;
      t = {Fmf.x * rc + Fmf.y * rs, Fmf.y * rc - Fmf.x * rs}; Fmf = t;
      t = {Fpb.x * rc - Fpb.y * rs, Fpb.x * rs + Fpb.y * rc}; Fpb = t;
      t = {Fmb.x * rc + Fmb.y * rs, Fmb.y * rc - Fmb.x * rs}; Fmb = t;
    }

    // ---- RF pulse (free pool only) ----
    const float a2 = 0.5f * fa * B1;
    const float c  = __cosf(a2), sn = __sinf(a2);
    const float c2 = c * c, s2 = sn * sn, cs = c * sn, czms = c2 - s2;
    const float cb = __cosf(ph), snb = __sinf(ph);
    const float cb2 = cb * cb - snb * snb, sb2 = 2.f * cb * snb;

    const float2 F = Fpf, G = Fmf, Z = Zf;
    const float2 Zei  = {Z.x * cb - Z.y * snb, Z.x * snb + Z.y * cb};   // Z*e^{i b}
    const float2 Zeim = {Z.x * cb + Z.y * snb, Z.y * cb - Z.x * snb};   // Z*e^{-i b}
    const float2 cGe2  = {G.x * cb2 + G.y * sb2,  G.x * sb2 - G.y * cb2};  // conj(G)*e^{2ib}
    const float2 cFe2m = {F.x * cb2 - F.y * sb2, -F.x * sb2 - F.y * cb2};  // conj(F)*e^{-2ib}
    const float2 Fpn = {c2 * F.x + s2 * cGe2.x - cs * Zei.y,
                        c2 * F.y + s2 * cGe2.y + cs * Zei.x};
    const float2 Fmn = {s2 * cFe2m.x + c2 * G.x + cs * Zeim.y,
                        s2 * cFe2m.y + c2 * G.y - cs * Zeim.x};
    const float2 Fem = {F.x * cb + F.y * snb, F.y * cb - F.x * snb};    // F*e^{-i b}
    const float2 Gep = {G.x * cb - G.y * snb, G.x * snb + G.y * cb};    // G*e^{i b}
    const float2 d   = {Fem.x - Gep.x, Fem.y - Gep.y};
    Zf = {cs * d.y + czms * Z.x, -cs * d.x + czms * Z.y};               // -i*cs*d + (c2-s2)*Z

    // ---- gradient shift across the 128 states (spans 4 waves -> LDS) ----
    const int q2 = p & 1;
    rollb[q2][0][s] = Fpn;
    rollb[q2][1][s] = Fmn;
    rollb[q2][2][s] = Fpb;
    rollb[q2][3][s] = Fmb;
    __syncthreads();
    const int sm = (s == 0) ? 0 : s - 1;
    const int sp = (s == N_STATES - 1) ? s : s + 1;
    const float2 z0 = {0.f, 0.f};
    Fpf = (s == 0)            ? z0 : rollb[q2][0][sm];
    Fmf = (s == N_STATES - 1) ? z0 : rollb[q2][1][sp];
    Fpb = (s == 0)            ? z0 : rollb[q2][2][sm];
    Fmb = (s == N_STATES - 1) ? z0 : rollb[q2][3][sp];

    // ---- stage post-shift carry for this pulse ----
    const int bq   = p >> 1;        // batch index (2 pulses per batch)
    const int bsel = bq % 3;        // staging buffer
    const int pib  = p & 1;         // pulse within batch
    stg[bsel][0][pib][s] = Fpf;
    stg[bsel][1][pib][s] = Fmf;
    stg[bsel][2][pib][s] = Zf;
    stg[bsel][3][pib][s] = Fpb;
    stg[bsel][4][pib][s] = Fmb;
    stg[bsel][5][pib][s] = Zb;

    // ---- wave0 drains the previous (completed) batch via the TDM ----
    // Staging writes of batch bq-1 are ordered before this by the barrier above.
    if (wave0 && (p & 1) == 0 && p >= 2) {
      const int qb = bq - 1;
      const uint32_t lds = (uint32_t)(uintptr_t)&stg[qb % 3][0][0][0];
      tdm_store_batch(lds, outb + (uint64_t)qb * 2048u);
    }
  }

  // ---- drain final batch ----
  __syncthreads();
  if (wave0) {
    const int qb = (N_PULSES / 2) - 1;
    const uint32_t lds = (uint32_t)(uintptr_t)&stg[qb % 3][0][0][0];
    tdm_store_batch(lds, outb + (uint64_t)qb * 2048u);
    __builtin_amdgcn_s_wait_tensorcnt(0);
  }
}

extern "C" void kernel_launch(void* const* d_in, const int* in_sizes, int n_in,
                              void* d_out, int out_size, void* d_ws, size_t ws_size,
                              hipStream_t stream) {
  (void)in_sizes; (void)n_in; (void)out_size; (void)d_ws; (void)ws_size;
  const float* fa  = (const float*)d_in[0];
  const float* ph  = (const float*)d_in[1];
  const float* T1f = (const float*)d_in[2];
  const float* T2f = (const float*)d_in[3];
  const float* T1b = (const float*)d_in[4];
  const float* T2b = (const float*)d_in[5];
  const float* kf  = (const float*)d_in[6];
  const float* kb  = (const float*)d_in[7];
  const float* TR  = (const float*)d_in[8];
  // d_in[9] = TE (unused by the reference math)
  const float* B0  = (const float*)d_in[10];
  const float* B1  = (const float*)d_in[11];
  const float* wf  = (const float*)d_in[12];
  const float* wb  = (const float*)d_in[13];

  epg_mt_kernel<<<dim3(1), dim3(128), 0, stream>>>(
      fa, ph, T1f, T2f, T1b, T2b, kf, kb, TR, B0, B1, wf, wb, (float*)d_out);
}